// CombinedLoss_44796508897917
// MI455X (gfx1250) — compile-verified
//
#include <hip/hip_runtime.h>
#include <hip/hip_bf16.h>
#include <math.h>

typedef __attribute__((ext_vector_type(2))) float v2f;
typedef __attribute__((ext_vector_type(8))) float v8f;

#define HD_H 512
#define HD_W 512
#define KTILE 32
#define TSTRIDE 36   // 32 + 4 pad: (36*l) mod 64 distinct for l=0..15 -> conflict-free column reads
#define SMOOTH 1e-5f
#define INF_BITS 0x7f800000u

// ---------------------------------------------------------------------------
// Kernel 1: squared row norms per (b,c): ws_x2[bc][i] = ||x_row_i||^2, same y.
// ---------------------------------------------------------------------------
__global__ __launch_bounds__(256) void norms_kernel(
    const float* __restrict__ x, const float* __restrict__ y,
    float* __restrict__ x2, float* __restrict__ y2)
{
    const int bc = blockIdx.x;                       // 0..95
    const float* xb = x + (size_t)bc * HD_H * HD_W;
    const float* yb = y + (size_t)bc * HD_H * HD_W;
    const int wave = threadIdx.x >> 5, lane = threadIdx.x & 31;

    for (int job = wave; job < 2 * HD_H; job += 8) {
        const float* row = (job < HD_H) ? (xb + (size_t)job * HD_W)
                                        : (yb + (size_t)(job - HD_H) * HD_W);
        float s = 0.0f;
        for (int k = lane; k < HD_W; k += 32) { float v = row[k]; s += v * v; }
        for (int off = 16; off > 0; off >>= 1) s += __shfl_xor(s, off, 32);
        if (lane == 0) {
            if (job < HD_H) x2[bc * HD_H + job] = s;
            else            y2[bc * HD_H + job - HD_H] = s;
        }
    }
}

// ---------------------------------------------------------------------------
// Kernel 2: Hausdorff min pass. Grid (16, 96): blockIdx.y = (b,c) slice,
// blockIdx.x = 128x128 supertile (si,sj). Cooperative GEMM: the workgroup
// stages 128xKTILE tiles of x and y in LDS per chunk; each of the 8 waves
// register-blocks a 64x32 patch (4x2 accumulators) of the x.y^T Gram matrix
// with V_WMMA_F32_16X16X4_F32, then folds d2 = x2[i]+y2[j]-2*dot into
// block-local row/col mins (LDS atomic_min_u32 on non-negative-float bits),
// flushed with one global atomicMin per row/col.
// ---------------------------------------------------------------------------
__global__ __launch_bounds__(256) void hd_min_kernel(
    const float* __restrict__ x, const float* __restrict__ y,
    const float* __restrict__ x2, const float* __restrict__ y2,
    unsigned* __restrict__ mins /* [96][1024]: 512 rowmin then 512 colmin */)
{
    const int bc = blockIdx.y;                       // 0..95
    const int st = blockIdx.x;                       // 0..15
    const int si = st >> 2, sj = st & 3;
    const int r0 = si * 128, c0 = sj * 128;          // supertile origin
    const float* xb = x + (size_t)bc * HD_H * HD_W;
    const float* yb = y + (size_t)bc * HD_H * HD_W;

    __shared__ __align__(16) float s_xt[128 * TSTRIDE];
    __shared__ __align__(16) float s_yt[128 * TSTRIDE];
    __shared__ float    s_x2l[128];
    __shared__ float    s_y2l[128];
    __shared__ unsigned s_min2[256];                 // [0,128): rowmin, [128,256): colmin

    const int tid  = threadIdx.x;
    const int wave = tid >> 5;
    const int lane = tid & 31;
    const int wm   = wave >> 2;                      // 0..1 -> row offset wm*64
    const int wn   = wave & 3;                       // 0..3 -> col offset wn*32
    const int half = lane >> 4;                      // frag half select
    const int mrow = lane & 15;

    if (tid < 128) {
        s_x2l[tid] = x2[bc * HD_H + r0 + tid];
        s_y2l[tid] = y2[bc * HD_H + c0 + tid];
    }
    s_min2[tid] = INF_BITS;                          // tid in [0,256)

    v8f acc[4][2];
    #pragma unroll
    for (int mi = 0; mi < 4; ++mi)
        #pragma unroll
        for (int nj = 0; nj < 2; ++nj)
            acc[mi][nj] = (v8f)(0.0f);

    for (int k0 = 0; k0 < HD_W; k0 += KTILE) {
        __syncthreads();                             // prior compute done / init visible
        // ---- cooperative stage: 128x32 of x rows r0.., y rows c0.. ----
        #pragma unroll
        for (int r = 0; r < 4; ++r) {
            int e   = tid + 256 * r;                 // 0..1023 float4 slots
            int row = e >> 3;                        // 0..127
            int c4  = (e & 7) << 2;                  // 0..28 step 4
            float4 vx = *(const float4*)(xb + (size_t)(r0 + row) * HD_W + k0 + c4);
            float4 vy = *(const float4*)(yb + (size_t)(c0 + row) * HD_W + k0 + c4);
            *(float4*)(&s_xt[row * TSTRIDE + c4]) = vx;
            *(float4*)(&s_yt[row * TSTRIDE + c4]) = vy;
        }
        __syncthreads();
        // ---- compute: 8 k-steps x 8 WMMA; A frags shared across nj, B across mi ----
        #pragma unroll
        for (int kk = 0; kk < KTILE; kk += 4) {
            v2f a[4], b[2];
            #pragma unroll
            for (int mi = 0; mi < 4; ++mi) {
                const float* p = &s_xt[(wm * 64 + mi * 16 + mrow) * TSTRIDE + kk + 2 * half];
                a[mi].x = p[0]; a[mi].y = p[1];
            }
            #pragma unroll
            for (int nj = 0; nj < 2; ++nj) {
                const float* p = &s_yt[(wn * 32 + nj * 16 + mrow) * TSTRIDE + kk + 2 * half];
                b[nj].x = p[0]; b[nj].y = p[1];
            }
            #pragma unroll
            for (int mi = 0; mi < 4; ++mi)
                #pragma unroll
                for (int nj = 0; nj < 2; ++nj)
                    acc[mi][nj] = __builtin_amdgcn_wmma_f32_16x16x4_f32(
                        false, a[mi], false, b[nj], (short)0, acc[mi][nj], false, false);
        }
    }

    // ---- epilogue: d2 and block-local row/col mins ----
    // C/D layout: VGPR r -> M = r + 8*half, N = lane&15.
    const int n = lane & 15;
    #pragma unroll
    for (int mi = 0; mi < 4; ++mi) {
        const int m0 = wm * 64 + mi * 16;
        #pragma unroll
        for (int nj = 0; nj < 2; ++nj) {
            const int n0 = wn * 32 + nj * 16;
            const float y2v = s_y2l[n0 + n];
            #pragma unroll
            for (int r = 0; r < 8; ++r) {
                const int m = m0 + r + half * 8;
                float d2 = fmaxf(s_x2l[m] + y2v - 2.0f * acc[mi][nj][r], 0.0f);
                unsigned bits = __float_as_uint(d2);   // order-preserving for d2 >= 0
                atomicMin(&s_min2[m], bits);
                atomicMin(&s_min2[128 + n0 + n], bits);
            }
        }
    }
    __syncthreads();

    // ---- flush: one global atomic_min_u32 per supertile row/col ----
    unsigned* mbc = mins + (size_t)bc * 1024;
    if (tid < 128) atomicMin(&mbc[r0 + tid], s_min2[tid]);
    else           atomicMin(&mbc[512 + c0 + (tid - 128)], s_min2[tid]);
}

// ---------------------------------------------------------------------------
// Kernel 3: per-(b,c) hd = sqrt(max over 1024 row/col min-d2 entries).
// ---------------------------------------------------------------------------
__global__ __launch_bounds__(256) void hd_reduce_kernel(
    const unsigned* __restrict__ mins, float* __restrict__ hd_out)
{
    const int bc = blockIdx.x;
    const unsigned* mbc = mins + (size_t)bc * 1024;
    __shared__ float s_red[8];
    const int tid = threadIdx.x, wave = tid >> 5, lane = tid & 31;

    unsigned mx = 0u;
    for (int i = tid; i < 1024; i += 256) { unsigned v = mbc[i]; mx = mx > v ? mx : v; }
    for (int off = 16; off > 0; off >>= 1) {
        unsigned o = (unsigned)__shfl_xor((int)mx, off, 32);
        mx = mx > o ? mx : o;
    }
    if (lane == 0) s_red[wave] = __uint_as_float(mx);
    __syncthreads();
    if (tid == 0) {
        unsigned m2 = 0u;
        for (int w = 0; w < 8; ++w) {
            unsigned v = __float_as_uint(s_red[w]);
            m2 = m2 > v ? m2 : v;
        }
        hd_out[bc] = sqrtf(__uint_as_float(m2));
    }
}

// ---------------------------------------------------------------------------
// Dice partial reduction: per-batch Σxy, Σx, Σy (float4 streaming).
// ---------------------------------------------------------------------------
__global__ __launch_bounds__(256) void dice_kernel(
    const float* __restrict__ x, const float* __restrict__ y,
    float* __restrict__ acc /* [32][3] */)
{
    const int b  = blockIdx.y;
    const int Nv = (3 * HD_H * HD_W) / 4;            // 196608 float4 per batch
    const float4* xb = (const float4*)x + (size_t)b * Nv;
    const float4* yb = (const float4*)y + (size_t)b * Nv;

    float si = 0.0f, sx = 0.0f, sy = 0.0f;
    for (int i = blockIdx.x * blockDim.x + threadIdx.x; i < Nv;
         i += gridDim.x * blockDim.x) {
        float4 a = xb[i], c = yb[i];
        si += a.x * c.x + a.y * c.y + a.z * c.z + a.w * c.w;
        sx += a.x + a.y + a.z + a.w;
        sy += c.x + c.y + c.z + c.w;
    }
    for (int off = 16; off > 0; off >>= 1) {
        si += __shfl_xor(si, off, 32);
        sx += __shfl_xor(sx, off, 32);
        sy += __shfl_xor(sy, off, 32);
    }
    __shared__ float p[3][8];
    const int wave = threadIdx.x >> 5, lane = threadIdx.x & 31;
    if (lane == 0) { p[0][wave] = si; p[1][wave] = sx; p[2][wave] = sy; }
    __syncthreads();
    if (threadIdx.x == 0) {
        float a0 = 0.0f, a1 = 0.0f, a2 = 0.0f;
        for (int w = 0; w < 8; ++w) { a0 += p[0][w]; a1 += p[1][w]; a2 += p[2][w]; }
        atomicAdd(&acc[b * 3 + 0], a0);
        atomicAdd(&acc[b * 3 + 1], a1);
        atomicAdd(&acc[b * 3 + 2], a2);
    }
}

__global__ void init_kernel(float* __restrict__ dice, unsigned* __restrict__ mins)
{
    const int i = blockIdx.x * blockDim.x + threadIdx.x;
    if (i < 96) dice[i] = 0.0f;
    if (i < 96 * 1024) mins[i] = INF_BITS;
}

__global__ void final_kernel(const float* __restrict__ dacc,
                             const float* __restrict__ hdv,
                             float* __restrict__ out)
{
    __shared__ float sh[128], sd[128];
    const int t = threadIdx.x;
    float h = (t < 96) ? hdv[t] : 0.0f;
    float d = 0.0f;
    if (t < 32) {
        float inter = dacc[t * 3 + 0];
        float sx    = dacc[t * 3 + 1];
        float sy    = dacc[t * 3 + 2];
        d = 1.0f - (2.0f * inter + SMOOTH) / (sx + sy + SMOOTH);
    }
    sh[t] = h; sd[t] = d;
    __syncthreads();
    if (t == 0) {
        float hs = 0.0f, ds = 0.0f;
        for (int i = 0; i < 128; ++i) { hs += sh[i]; ds += sd[i]; }
        out[0] = 0.4f * (ds / 32.0f) + 0.6f * (hs / 96.0f);
    }
}

// ---------------------------------------------------------------------------
extern "C" void kernel_launch(void* const* d_in, const int* in_sizes, int n_in,
                              void* d_out, int out_size, void* d_ws, size_t ws_size,
                              hipStream_t stream)
{
    const float* x = (const float*)d_in[0];
    const float* y = (const float*)d_in[1];
    float* out     = (float*)d_out;

    // Workspace layout (float units):
    //   [0,96)                 dice accumulators [B][inter,sumx,sumy]
    //   [128,224)              hd per (b,c)
    //   [256, 256+49152)       x2 norms [96][512]
    //   [49408, 49408+49152)   y2 norms [96][512]
    //   [98560, 98560+98304)   row/col min-d2 bits [96][1024] (as unsigned)
    float*    ws_dice = (float*)d_ws;
    float*    ws_hd   = (float*)d_ws + 128;
    float*    ws_x2   = (float*)d_ws + 256;
    float*    ws_y2   = (float*)d_ws + 49408;
    unsigned* ws_min  = (unsigned*)d_ws + 98560;

    init_kernel<<<(96 * 1024 + 255) / 256, 256, 0, stream>>>(ws_dice, ws_min);
    dice_kernel<<<dim3(64, 32), 256, 0, stream>>>(x, y, ws_dice);
    norms_kernel<<<96, 256, 0, stream>>>(x, y, ws_x2, ws_y2);
    hd_min_kernel<<<dim3(16, 96), 256, 0, stream>>>(x, y, ws_x2, ws_y2, ws_min);
    hd_reduce_kernel<<<96, 256, 0, stream>>>(ws_min, ws_hd);
    final_kernel<<<1, 128, 0, stream>>>(ws_dice, ws_hd, out);
}